// ViTBeans_57174604644752
// MI455X (gfx1250) — compile-verified
//
#include <hip/hip_runtime.h>
#include <math.h>

#define NE 16
#define D_FULL 1024
#define DMV 128
#define SS 64
#define HH 16
#define PP 2048
#define BB 8
#define CC 256
#define WW 3

typedef __attribute__((ext_vector_type(2))) float v2f;
typedef __attribute__((ext_vector_type(8))) float v8f;

#if __has_builtin(__builtin_amdgcn_global_load_async_to_lds_b128) && \
    __has_builtin(__builtin_amdgcn_s_wait_asynccnt)
#define HAVE_ASYNC_LDS 1
// Match the builtin's parameter type exactly as printed by clang:
//   "__attribute__((__vector_size__(4 * sizeof(int)))) int __device__ *"
typedef int v4i_g __attribute__((vector_size(16)));
typedef __attribute__((address_space(1))) v4i_g* as1_v4i;
typedef __attribute__((address_space(3))) v4i_g* as3_v4i;
#else
#define HAVE_ASYNC_LDS 0
#endif

__device__ __forceinline__ v8f wmma4(v2f a, v2f b, v8f c) {
  // D = A(16x4 f32) * B(4x16 f32) + C(16x16 f32)
  return __builtin_amdgcn_wmma_f32_16x16x4_f32(false, a, false, b, (short)0, c,
                                               false, false);
}

// ---------------- zero output ----------------
__global__ void k_zero(float* p, int n) {
  int i = blockIdx.x * blockDim.x + threadIdx.x;
  int stride = gridDim.x * blockDim.x;
  for (; i < n; i += stride) p[i] = 0.0f;
}

// ---------------- expert routes (Cantor coords, stable 3-NN) ----------------
__global__ void k_routes(int* routes) {
  int e = threadIdx.x;
  if (e >= NE) return;
  double co[NE];
  for (int i = 0; i < NE; i++) {
    double x = (double)i / (double)(NE - 1);
    x = fmax(1e-6, fmin(x, 1.0 - 1e-6));
    double val = 0.0, f = 0.5;
    for (int d = 0; d < 8; d++) {
      x *= 3.0;
      int dd = (int)x;
      x -= (double)dd;
      if (dd == 2) val += f;
      f *= 0.5;
    }
    co[i] = val;
  }
  bool used[NE];
  for (int i = 0; i < NE; i++) used[i] = false;
  for (int w = 0; w < WW; w++) {
    int best = 0;
    double bd = 1e30;
    for (int i = 0; i < NE; i++) {
      if (used[i]) continue;
      double d = fabs(co[i] - co[e]);
      if (d < bd) { bd = d; best = i; }  // ascending i => stable ties
    }
    used[best] = true;
    routes[e * WW + w] = best;
  }
}

// ---------------- dispatch: per-expert capacity-C compaction ----------------
__global__ void k_dispatch(const float* fp, int* idx, float* vld) {
  int e = blockIdx.x;
  int lane = threadIdx.x;  // 32 threads, one wave
  __shared__ int cnt;
  if (lane == 0) cnt = 0;
  __syncthreads();
  double lo = (double)e / NE, hi = (double)(e + 1) / NE;
  for (int base = 0; base < PP; base += 32) {
    int p = base + lane;
    double f = (double)fp[p];
    bool m = (f >= lo) && ((e == NE - 1) ? (f <= hi) : (f < hi));
    unsigned long long bal = __ballot(m);
    int pre = __popcll(bal & ((1ull << lane) - 1ull));
    int c = cnt;
    if (m && (c + pre) < CC) {
      idx[e * CC + c + pre] = p;
      vld[e * CC + c + pre] = 1.0f;
    }
    __syncthreads();
    if (lane == 0) cnt = min(CC, c + (int)__popcll(bal));
    __syncthreads();
  }
  for (int base = 0; base < PP && cnt < CC; base += 32) {
    int p = base + lane;
    double f = (double)fp[p];
    bool m = (f >= lo) && ((e == NE - 1) ? (f <= hi) : (f < hi));
    bool inv = !m;
    unsigned long long bal = __ballot(inv);
    int pre = __popcll(bal & ((1ull << lane) - 1ull));
    int c = cnt;
    if (inv && (c + pre) < CC) {
      idx[e * CC + c + pre] = p;
      vld[e * CC + c + pre] = 0.0f;
    }
    __syncthreads();
    if (lane == 0) cnt = min(CC, c + (int)__popcll(bal));
    __syncthreads();
  }
}

// ---------------- qd/kd = {q,k}_w @ normalized(penta)^T ----------------
__global__ void k_projdirs(const float* qw, const float* kw, const float* penta,
                           float* qd, float* kd) {
  int e = blockIdx.x;
  __shared__ float rn[5];
  int t = threadIdx.x;  // 320 threads: v = t/64, s = t%64
  if (t < 5) {
    const float* pv = penta + (size_t)(e * 5 + t) * DMV;
    float s = 0.0f;
    for (int d = 0; d < DMV; d++) s += pv[d] * pv[d];
    rn[t] = rsqrtf(s);
  }
  __syncthreads();
  int v = t >> 6, s = t & 63;
  const float* pv = penta + (size_t)(e * 5 + v) * DMV;
  const float* qr = qw + (size_t)(e * SS + s) * DMV;
  const float* kr = kw + (size_t)(e * SS + s) * DMV;
  float r = rn[v], aq = 0.0f, ak = 0.0f;
  for (int d = 0; d < DMV; d++) {
    float dir = pv[d] * r;
    aq += qr[d] * dir;
    ak += kr[d] * dir;
  }
  qd[(e * 5 + v) * SS + s] = aq;
  kd[(e * 5 + v) * SS + s] = ak;
}

// ---------------- per-(e,b): gather + gate + V GEMM (WMMA) + affinities ----
__global__ void k_expert(const float* tokens, const float* alpha,
                         const float* gw1, const float* gb1, const float* gw2,
                         const float* gb2, const float* vw, const float* qd,
                         const float* kd, const int* idx, float* Vout,
                         float* Qaff, float* Kaff) {
  int e = blockIdx.x, b = blockIdx.y;
  int tid = threadIdx.x;  // 256 threads
  __shared__ float feats[CC][SS + 1];  // +1 pad: avoid bank conflicts
  __shared__ float vwL[SS][DMV];       // staged B operand (32 KB, reused 16x)
  __shared__ int idxL[CC];
  idxL[tid] = idx[e * CC + tid];
  // stage v_w[e] (64x128) into LDS cooperatively: 2048 float4s / 256 threads
  {
    const float4* src = (const float4*)(vw + (size_t)e * SS * DMV);
    float4* dst = (float4*)&vwL[0][0];
    for (int i = tid; i < SS * DMV / 4; i += 256) dst[i] = src[i];
  }
  // gather my row (256 rows, one per thread)
  {
    const float* src =
        tokens + ((size_t)b * PP + (size_t)idxL[tid]) * D_FULL + e * SS;
    for (int s = 0; s < SS; s++) feats[tid][s] = src[s];
  }
  // gate MLP + scale + affinity projections (row-local)
  {
    const float* w1 = gw1 + e * SS * HH;
    const float* b1 = gb1 + e * HH;
    const float* w2 = gw2 + e * HH;
    float acc2 = gb2[e];
    for (int j = 0; j < HH; j++) {
      float a = b1[j];
      for (int s = 0; s < SS; s++) a += feats[tid][s] * w1[s * HH + j];
      float t = 0.7978845608028654f * (a + 0.044715f * a * a * a);
      float g = 0.5f * a * (1.0f + tanhf(t));
      acc2 += g * w2[j];
    }
    float gate = 1.0f / (1.0f + __expf(-acc2));
    float aw = 1.0f / (1.0f + __expf(-alpha[e]));
    float sc = gate * aw + (1.0f - aw);
    for (int s = 0; s < SS; s++) feats[tid][s] *= sc;
    for (int v = 0; v < 5; v++) {
      const float* qv = qd + (e * 5 + v) * SS;
      const float* kv = kd + (e * 5 + v) * SS;
      float aq = 0.0f, ak = 0.0f;
      for (int s = 0; s < SS; s++) {
        float f = feats[tid][s];
        aq += f * qv[s];
        ak += f * kv[s];
      }
      Qaff[((size_t)(e * BB + b) * 5 + v) * CC + tid] = aq;
      Kaff[((size_t)(e * BB + b) * 5 + v) * CC + tid] = ak;
    }
  }
  __syncthreads();
  // V = feats(256x64) @ vw[e](64x128) via V_WMMA_F32_16X16X4_F32
  int lane = tid & 31, wave = tid >> 5;
  int hl = lane & 15;
  bool hih = lane >= 16;
  float* vo = Vout + (size_t)(e * BB + b) * CC * DMV;
  for (int tile = wave; tile < 16 * 8; tile += 8) {
    int mt = tile >> 3, nt = tile & 7;
    v8f acc = {};
    int row = mt * 16 + hl;
    int col = nt * 16 + hl;
#pragma unroll
    for (int k = 0; k < SS; k += 4) {
      int kk = k + (hih ? 2 : 0);
      v2f a, bb;
      a.x = feats[row][kk];
      a.y = feats[row][kk + 1];
      bb.x = vwL[kk][col];
      bb.y = vwL[kk + 1][col];
      acc = wmma4(a, bb, acc);
    }
    int rbase = mt * 16 + (hih ? 8 : 0);
#pragma unroll
    for (int r = 0; r < 8; r++) vo[(rbase + r) * DMV + col] = acc[r];
  }
}

// ---------------- per-(e,b): rank-1 attention + attn@V (WMMA) -------------
__global__ void k_attn(const float* Vws, const float* QaffW, const float* KaffW,
                       const int* idx, const float* vld, const int* routes,
                       const float* fusion_w, const float* temperature,
                       float* out) {
  int e = blockIdx.x, b = blockIdx.y;
  int tid = threadIdx.x;  // 256 threads
  int lane = tid & 31, wave = tid >> 5;
  __shared__ float ka[5][WW * CC];
  __shared__ float vmnb[WW * CC];
  __shared__ float qa[5][CC];
  __shared__ float mrow[5][CC];
  __shared__ float sfac[5][CC];
  __shared__ float Vch[32][DMV];
  __shared__ int idxE[CC];
  __shared__ float vmE[CC];
  __shared__ float wdirS[5];
  __shared__ int nbS[WW];

  if (tid < WW) nbS[tid] = routes[e * WW + tid];
  if (tid == 0) {
    float mx = fusion_w[0];
    for (int v = 1; v < 5; v++) mx = fmaxf(mx, fusion_w[v]);
    float s = 0.0f, ex[5];
    for (int v = 0; v < 5; v++) { ex[v] = __expf(fusion_w[v] - mx); s += ex[v]; }
    for (int v = 0; v < 5; v++) wdirS[v] = ex[v] / s;
  }
  idxE[tid] = idx[e * CC + tid];
  vmE[tid] = vld[e * CC + tid];
  __syncthreads();
  for (int n = 0; n < WW; n++) {
    int ne = nbS[n];
    vmnb[n * CC + tid] = vld[ne * CC + tid];
    for (int v = 0; v < 5; v++)
      ka[v][n * CC + tid] = KaffW[((size_t)(ne * BB + b) * 5 + v) * CC + tid];
  }
  for (int v = 0; v < 5; v++)
    qa[v][tid] = QaffW[((size_t)(e * BB + b) * 5 + v) * CC + tid];
  __syncthreads();
  float invt = 1.0f / temperature[0];
  // pass 1: per-(v,row) softmax stats over 768 masked rank-1 scores
  for (int v = 0; v < 5; v++) {
    float q = qa[v][tid] * invt;
    float m = -3.0e38f;
    for (int j = 0; j < WW * CC; j++)
      m = (vmnb[j] > 0.0f) ? fmaxf(m, q * ka[v][j]) : m;
    if (m < -1.0e38f) m = 0.0f;
    float d = 0.0f;
    for (int j = 0; j < WW * CC; j++)
      d += (vmnb[j] > 0.0f) ? __expf(q * ka[v][j] - m) : 0.0f;
    mrow[v][tid] = m;
    sfac[v][tid] = wdirS[v] / fmaxf(d, 1e-30f);
  }
  __syncthreads();
  // pass 2: out[i,:] = sum_v sum_j P'[i,j] * V[j,:]   (K folded over v: 5*768)
  int hl = lane & 15;
  bool hih = lane >= 16;
  v8f acc[2][8];
#pragma unroll
  for (int m2 = 0; m2 < 2; m2++)
#pragma unroll
    for (int nt = 0; nt < 8; nt++) acc[m2][nt] = (v8f){};
  for (int v = 0; v < 5; v++) {
    float qrow[2], mr[2], sf[2];
#pragma unroll
    for (int m2 = 0; m2 < 2; m2++) {
      int row = (wave * 2 + m2) * 16 + hl;
      qrow[m2] = qa[v][row] * invt;
      mr[m2] = mrow[v][row];
      sf[m2] = sfac[v][row];
    }
    for (int ch = 0; ch < (WW * CC) / 32; ch++) {  // 24 chunks of 32 keys
      __syncthreads();
      // stage V chunk (32 rows x 128 f32 = 16 KB) into LDS
      for (int r4 = tid; r4 < 32 * (DMV / 4); r4 += 256) {
        int r = r4 / (DMV / 4);
        int c4 = r4 % (DMV / 4);
        int j = ch * 32 + r;
        int n = j >> 8, jj = j & 255;
        const float4* src =
            (const float4*)(Vws + ((size_t)(nbS[n] * BB + b) * CC + jj) * DMV);
#if HAVE_ASYNC_LDS
        // gfx1250 async copy engine: GLOBAL_LOAD_ASYNC_TO_LDS_B128 (ASYNCcnt)
        __builtin_amdgcn_global_load_async_to_lds_b128(
            (as1_v4i)(void*)(src + c4), (as3_v4i)(void*)&Vch[r][c4 * 4], 0, 0);
#else
        ((float4*)Vch[r])[c4] = src[c4];
#endif
      }
#if HAVE_ASYNC_LDS
      __builtin_amdgcn_s_wait_asynccnt(0);
#endif
      __syncthreads();
#pragma unroll
      for (int ks = 0; ks < 8; ks++) {
        int kloc = ks * 4 + (hih ? 2 : 0);
        int jg = ch * 32 + kloc;
        v2f afr[2];
#pragma unroll
        for (int m2 = 0; m2 < 2; m2++) {
          float p0 = (vmnb[jg] > 0.0f)
                         ? sf[m2] * __expf(qrow[m2] * ka[v][jg] - mr[m2])
                         : 0.0f;
          float p1 = (vmnb[jg + 1] > 0.0f)
                         ? sf[m2] * __expf(qrow[m2] * ka[v][jg + 1] - mr[m2])
                         : 0.0f;
          afr[m2].x = p0;
          afr[m2].y = p1;
        }
#pragma unroll
        for (int nt = 0; nt < 8; nt++) {
          int col = nt * 16 + hl;
          v2f bfr;
          bfr.x = Vch[kloc][col];
          bfr.y = Vch[kloc + 1][col];
          acc[0][nt] = wmma4(afr[0], bfr, acc[0][nt]);
          acc[1][nt] = wmma4(afr[1], bfr, acc[1][nt]);
        }
      }
    }
  }
  __syncthreads();
  // scatter valid rows (each valid (b,p) owned by exactly one (e,i))
#pragma unroll
  for (int m2 = 0; m2 < 2; m2++) {
    int rbase = (wave * 2 + m2) * 16 + (hih ? 8 : 0);
#pragma unroll
    for (int r = 0; r < 8; r++) {
      int row = rbase + r;
      if (vmE[row] > 0.0f) {
        float* dst = out + ((size_t)b * PP + (size_t)idxE[row]) * DMV;
#pragma unroll
        for (int nt = 0; nt < 8; nt++) dst[nt * 16 + hl] = acc[m2][nt][r];
      }
    }
  }
}

extern "C" void kernel_launch(void* const* d_in, const int* in_sizes, int n_in,
                              void* d_out, int out_size, void* d_ws,
                              size_t ws_size, hipStream_t stream) {
  const float* tokens = (const float*)d_in[0];
  const float* fingerprints = (const float*)d_in[1];
  const float* alpha = (const float*)d_in[2];
  const float* gate_w1 = (const float*)d_in[3];
  const float* gate_b1 = (const float*)d_in[4];
  const float* gate_w2 = (const float*)d_in[5];
  const float* gate_b2 = (const float*)d_in[6];
  const float* q_w = (const float*)d_in[7];
  const float* k_w = (const float*)d_in[8];
  const float* v_w = (const float*)d_in[9];
  const float* penta = (const float*)d_in[10];
  const float* fusion_w = (const float*)d_in[11];
  const float* temperature = (const float*)d_in[12];
  float* out = (float*)d_out;

  char* ws = (char*)d_ws;
  size_t off = 0;
  auto alloc = [&](size_t bytes) {
    void* p = ws + off;
    off = (off + bytes + 255) & ~(size_t)255;
    return p;
  };
  int* idx = (int*)alloc(NE * CC * sizeof(int));
  float* vld = (float*)alloc(NE * CC * sizeof(float));
  int* routes = (int*)alloc(NE * WW * sizeof(int));
  float* qd = (float*)alloc(NE * 5 * SS * sizeof(float));
  float* kd = (float*)alloc(NE * 5 * SS * sizeof(float));
  float* Qaff = (float*)alloc((size_t)NE * BB * 5 * CC * sizeof(float));
  float* Kaff = (float*)alloc((size_t)NE * BB * 5 * CC * sizeof(float));
  float* Vws = (float*)alloc((size_t)NE * BB * CC * DMV * sizeof(float));

  k_zero<<<2048, 256, 0, stream>>>(out, out_size);
  k_routes<<<1, 32, 0, stream>>>(routes);
  k_dispatch<<<NE, 32, 0, stream>>>(fingerprints, idx, vld);
  k_projdirs<<<NE, 320, 0, stream>>>(q_w, k_w, penta, qd, kd);
  k_expert<<<dim3(NE, BB), 256, 0, stream>>>(tokens, alpha, gate_w1, gate_b1,
                                             gate_w2, gate_b2, v_w, qd, kd, idx,
                                             Vws, Qaff, Kaff);
  k_attn<<<dim3(NE, BB), 256, 0, stream>>>(Vws, Qaff, Kaff, idx, vld, routes,
                                           fusion_w, temperature, out);
  (void)in_sizes;
  (void)n_in;
  (void)ws_size;
}